// DifferentiableCompressor_54571854463268
// MI455X (gfx1250) — compile-verified
//
#include <hip/hip_runtime.h>
#include <stdint.h>

// ---------------------------------------------------------------------------
// Differentiable compressor for MI455X (gfx1250).
//   pass 1: rect_db = 20*log10(|x|+1e-8)            (elementwise, f32x4)
//   pass 2: attack/release leaky integrator scan     (serial per row; LDS +
//           async global<->LDS movement, double buffered)
//   pass 3: knee gains + out = x * 10^(gain_db/20)   (elementwise, f32x4)
// ---------------------------------------------------------------------------

#define NLANES 32
#define CHUNK 2048                                   // floats per scan chunk
#define CHUNK_BYTES (CHUNK * 4)
#define AOPS ((CHUNK_BYTES) / (NLANES * 16))         // b128 async ops per chunk = 16

// Precomputed constants (double-precision derived, rounded to f32):
#define A_AT 0.99773499531f        // exp(-1/(10ms*44.1kHz/1000))
#define A_REL 0.99977349834f       // exp(-1/(100ms*44.1kHz/1000))
#define KNEE 0.1f
#define C_DN (-0.49250374815f)     // -(1 - 1/66.7) * 0.5
#define C_UP 0.45f                 // (1 - 0.1) * 0.5
#define UPMAX 36.0f
#define TMIN (-40.0f)
#define TSPAN 40.0f
#define DB_FROM_LOG2 6.0205999133f     // 20*log10(y) = this * log2(y)
#define LOG2_10_OVER_20 0.16609640474f // 10^(g/20) = exp2(g * this)

// Low 32 bits of a flat pointer into LDS == LDS byte address (HW adds LDS_BASE).
__device__ __forceinline__ unsigned lds_addr_of(const void* p) {
  return (unsigned)(uintptr_t)p;
}

// ---------------------------------------------------------------------------
// Pass 1: rect_db
// ---------------------------------------------------------------------------
__global__ void compressor_rectdb_kernel(const float* __restrict__ x,
                                         float* __restrict__ rdb, int N) {
  const int row = blockIdx.y;
  const int i4 = blockIdx.x * blockDim.x + threadIdx.x;
  const int n4 = N >> 2;
  if (i4 >= n4) return;
  const float4* xp = (const float4*)(x + (size_t)row * N);
  float4* rp = (float4*)(rdb + (size_t)row * N);
  float4 v = xp[i4];
  float4 r;
  r.x = DB_FROM_LOG2 * __builtin_log2f(fabsf(v.x) + 1e-8f);
  r.y = DB_FROM_LOG2 * __builtin_log2f(fabsf(v.y) + 1e-8f);
  r.z = DB_FROM_LOG2 * __builtin_log2f(fabsf(v.z) + 1e-8f);
  r.w = DB_FROM_LOG2 * __builtin_log2f(fabsf(v.w) + 1e-8f);
  rp[i4] = r;
}

// ---------------------------------------------------------------------------
// Pass 2: serial attack/release scan, one wave per row.
// In place: lvl[] holds rect_db on entry, level_db on exit.
// ---------------------------------------------------------------------------
__global__ __launch_bounds__(NLANES, 1) void compressor_scan_kernel(
    float* __restrict__ lvl, int N) {
  __shared__ float inbuf[2][CHUNK];   // double-buffered async-load target
  __shared__ float outbuf[2][CHUNK];  // double-buffered async-store source

  const int row = blockIdx.x;
  const int lane = threadIdx.x;
  float* rowp = lvl + (size_t)row * N;
  const int nfull = N / CHUNK;

  // Scan carry init = rect_db[row][0]; first step of the reference recurrence
  // with v == curr yields curr, so starting the loop at i=0 is consistent.
  float curr = rowp[0];

  // Prefetch chunk 0: 32 lanes x 16B per op, 16 ops -> 8 KB.
  if (nfull > 0) {
#pragma unroll
    for (int j = 0; j < AOPS; ++j) {
      unsigned lo = lds_addr_of(&inbuf[0][j * 128 + lane * 4]);
      unsigned go = (unsigned)(j * 512 + lane * 16);
      asm volatile("global_load_async_to_lds_b128 %0, %1, %2 offset:0" ::"v"(lo),
                   "v"(go), "s"(rowp)
                   : "memory");
    }
  }

  for (int c = 0; c < nfull; ++c) {
    const int cb = c & 1;
    if (c + 1 < nfull) {
      // Prefetch chunk c+1 into the other input buffer.
      const unsigned base = (unsigned)(c + 1) * CHUNK_BYTES;
#pragma unroll
      for (int j = 0; j < AOPS; ++j) {
        unsigned lo = lds_addr_of(&inbuf[(c + 1) & 1][j * 128 + lane * 4]);
        unsigned go = base + (unsigned)(j * 512 + lane * 16);
        asm volatile("global_load_async_to_lds_b128 %0, %1, %2 offset:0" ::"v"(
                         lo),
                     "v"(go), "s"(rowp)
                     : "memory");
      }
      // ASYNCcnt <= AOPS: only the just-issued 16 loads may remain. Loads
      // retire in order and stores retire in order, so this proves: load c
      // done, store c-1 possibly pending (different outbuf), store c-2 done
      // (same outbuf we are about to reuse).
      asm volatile("s_wait_asynccnt 16" ::: "memory");
    } else {
      asm volatile("s_wait_asynccnt 0" ::: "memory");
    }

    // Serial nonlinear recurrence — the true critical path. LDS-resident.
    if (lane == 0) {
#pragma unroll 4
      for (int i = 0; i < CHUNK; ++i) {
        float v = inbuf[cb][i];
        float a = (v > curr) ? A_AT : A_REL;
        curr = __builtin_fmaf(a, curr - v, v);  // a*curr + (1-a)*v
        outbuf[cb][i] = curr;
      }
    }

    // Ensure lane0's ds_stores landed in LDS before the async-store engine
    // reads them, then stream the level chunk back over rect_db (in place).
    asm volatile("s_wait_dscnt 0" ::: "memory");
    {
      const unsigned base = (unsigned)c * CHUNK_BYTES;
#pragma unroll
      for (int j = 0; j < AOPS; ++j) {
        unsigned lo = lds_addr_of(&outbuf[cb][j * 128 + lane * 4]);
        unsigned go = base + (unsigned)(j * 512 + lane * 16);
        asm volatile("global_store_async_from_lds_b128 %0, %1, %2 offset:0" ::
                         "v"(go),
                     "v"(lo), "s"(rowp)
                     : "memory");
      }
    }
  }

  // Tail (N mod CHUNK), synchronous path.
  const int rem = N - nfull * CHUNK;
  if (rem > 0) {
    asm volatile("s_wait_asynccnt 0" ::: "memory");
    const int base = nfull * CHUNK;
    for (int i = lane; i < rem; i += NLANES) inbuf[0][i] = rowp[base + i];
    __syncthreads();
    if (lane == 0) {
      for (int i = 0; i < rem; ++i) {
        float v = inbuf[0][i];
        float a = (v > curr) ? A_AT : A_REL;
        curr = __builtin_fmaf(a, curr - v, v);
        outbuf[0][i] = curr;
      }
    }
    __syncthreads();
    for (int i = lane; i < rem; i += NLANES) rowp[base + i] = outbuf[0][i];
  }
  asm volatile("s_wait_asynccnt 0" ::: "memory");
}

// ---------------------------------------------------------------------------
// Pass 3: knee gain + output
// ---------------------------------------------------------------------------
__device__ __forceinline__ float compressor_gain(float level, float tdb,
                                                 float d) {
  float dn = level - tdb;
  float tdn = dn - KNEE;
  float gdn =
      (dn > -KNEE) ? C_DN * (dn + KNEE + sqrtf(tdn * tdn + 1e-4f)) : 0.0f;
  float up = tdb - level;
  float tup = up - KNEE;
  float gup =
      (up > -KNEE)
          ? fminf(C_UP * (up + KNEE + sqrtf(tup * tup + 1e-4f)), UPMAX)
          : 0.0f;
  float gdb = (gdn + gup) * d;
  return __builtin_exp2f(gdb * LOG2_10_OVER_20);
}

__global__ void compressor_gain_kernel(const float* __restrict__ x,
                                       const float* __restrict__ lvl,
                                       const float* __restrict__ thr,
                                       const float* __restrict__ dep,
                                       float* __restrict__ out, int N) {
  const int row = blockIdx.y;
  const int i4 = blockIdx.x * blockDim.x + threadIdx.x;
  const int n4 = N >> 2;
  if (i4 >= n4) return;
  const float tdb = TMIN + thr[row] * TSPAN;
  const float d = dep[row];
  float4 L = ((const float4*)(lvl + (size_t)row * N))[i4];
  float4 X = ((const float4*)(x + (size_t)row * N))[i4];
  float4 O;
  O.x = X.x * compressor_gain(L.x, tdb, d);
  O.y = X.y * compressor_gain(L.y, tdb, d);
  O.z = X.z * compressor_gain(L.z, tdb, d);
  O.w = X.w * compressor_gain(L.w, tdb, d);
  ((float4*)(out + (size_t)row * N))[i4] = O;
}

// ---------------------------------------------------------------------------
extern "C" void kernel_launch(void* const* d_in, const int* in_sizes, int n_in,
                              void* d_out, int out_size, void* d_ws,
                              size_t ws_size, hipStream_t stream) {
  const float* x = (const float*)d_in[0];
  const float* thr = (const float*)d_in[1];
  const float* dep = (const float*)d_in[2];
  float* out = (float*)d_out;

  const int B = in_sizes[1];           // 16 rows
  const int N = in_sizes[0] / B;       // 441000 samples per row
  float* lvl = (float*)d_ws;           // B*N floats of scratch (28.2 MB)

  dim3 blk(256);
  dim3 grd(((N >> 2) + 255) / 256, B);
  compressor_rectdb_kernel<<<grd, blk, 0, stream>>>(x, lvl, N);
  compressor_scan_kernel<<<dim3(B), dim3(NLANES), 0, stream>>>(lvl, N);
  compressor_gain_kernel<<<grd, blk, 0, stream>>>(x, lvl, thr, dep, out, N);
}